// GNN_23819888624267
// MI455X (gfx1250) — compile-verified
//
#include <hip/hip_runtime.h>
#include <hip/hip_bf16.h>

typedef float v2f __attribute__((ext_vector_type(2)));
typedef float v8f __attribute__((ext_vector_type(8)));

// ---------------------------------------------------------------------------
// deg[i] = 1.0 (self-loop contribution)
// ---------------------------------------------------------------------------
__global__ void k_init_deg(float* __restrict__ deg, int N) {
    int i = blockIdx.x * blockDim.x + threadIdx.x;
    if (i < N) deg[i] = 1.0f;
}

// ---------------------------------------------------------------------------
// deg[dst[e]] += 1 for all 16M edges (4 edges/thread, int4 loads)
// ---------------------------------------------------------------------------
__global__ void k_deg_count(const int* __restrict__ dst, float* __restrict__ deg, int E) {
    int t = blockIdx.x * blockDim.x + threadIdx.x;
    int e0 = t * 4;
    if (e0 + 3 < E) {
        int4 d = *(const int4*)(dst + e0);
        unsafeAtomicAdd(&deg[d.x], 1.0f);
        unsafeAtomicAdd(&deg[d.y], 1.0f);
        unsafeAtomicAdd(&deg[d.z], 1.0f);
        unsafeAtomicAdd(&deg[d.w], 1.0f);
    } else {
        for (int e = e0; e < E; ++e) unsafeAtomicAdd(&deg[dst[e]], 1.0f);
    }
}

// ---------------------------------------------------------------------------
// dis[i] = rsqrt(deg[i]);  q[i] = dis[i]*x[i];  S1[i] = q[i] (self-loop init)
// degS1 aliases: read deg, overwrite with S1 (same index, same thread -> safe)
// ---------------------------------------------------------------------------
__global__ void k_node_q(float* degS1, float* __restrict__ dis,
                         float* __restrict__ q, const float* __restrict__ x, int N) {
    int i = blockIdx.x * blockDim.x + threadIdx.x;
    if (i < N) {
        float r  = rsqrtf(degS1[i]);   // deg >= 1 always (self-loops)
        float qv = r * x[i];
        dis[i]   = r;
        q[i]     = qv;
        degS1[i] = qv;                 // becomes S1
    }
}

// ---------------------------------------------------------------------------
// Pass 1: S1[dst] += q[src]  (rank-1 layer-1 messages: 1 atomic/edge, not 8)
// ---------------------------------------------------------------------------
__global__ void k_edge1(const int* __restrict__ src, const int* __restrict__ dst,
                        const float* __restrict__ q, float* __restrict__ S1, int E) {
    int t = blockIdx.x * blockDim.x + threadIdx.x;
    int e0 = t * 4;
    if (e0 + 3 < E) {
        int4 s = *(const int4*)(src + e0);
        int4 d = *(const int4*)(dst + e0);
        unsafeAtomicAdd(&S1[d.x], q[s.x]);
        unsafeAtomicAdd(&S1[d.y], q[s.y]);
        unsafeAtomicAdd(&S1[d.z], q[s.z]);
        unsafeAtomicAdd(&S1[d.w], q[s.w]);
    } else {
        for (int e = e0; e < E; ++e) unsafeAtomicAdd(&S1[dst[e]], q[src[e]]);
    }
}

// ---------------------------------------------------------------------------
// Node transform between the two message passes, 16 nodes per wave32:
//   a     = dis[i]*S1[i]
//   h1[k] = relu(a*W1[k] + b1[k])                      k = 0..7
//   g     = h1 @ W2   via V_WMMA_F32_16X16X4_F32 x2    (16x8 @ 8x2, N-padded)
//   p[i,c]  = dis[i]*g[c];  S2[i,c] = p[i,c]           (self-loop init)
// A layout (f32 16x4): lanes 0-15 -> K=0,1 ; lanes 16-31 -> K=2,3 ; M = lane%16
// D layout: VGPR v on lane (m,half) = D[v + 8*half][m]
// EXEC is all-1s through the WMMAs; only the final stores are guarded.
// ---------------------------------------------------------------------------
__global__ void k_node_wmma(const float* __restrict__ dis, const float* __restrict__ S1,
                            const float* __restrict__ W1, const float* __restrict__ b1,
                            const float* __restrict__ W2,  // [8,2] row-major
                            float* __restrict__ p, float* __restrict__ S2, int N) {
    int lane = threadIdx.x & 31;
    int wave = threadIdx.x >> 5;
    int m    = lane & 15;
    int half = lane >> 4;
    int base = (blockIdx.x * (blockDim.x >> 5) + wave) * 16;

    int idx = base + m;
    if (idx > N - 1) idx = N - 1;          // clamp for compute; stores guarded
    float a = dis[idx] * S1[idx];

    int k0 = 2 * half;                      // this lane's K rows in the A tile
    v2f A0, A1, B0, B1;
    A0.x = fmaxf(fmaf(a, W1[k0 + 0], b1[k0 + 0]), 0.0f);
    A0.y = fmaxf(fmaf(a, W1[k0 + 1], b1[k0 + 1]), 0.0f);
    A1.x = fmaxf(fmaf(a, W1[4 + k0 + 0], b1[4 + k0 + 0]), 0.0f);
    A1.y = fmaxf(fmaf(a, W1[4 + k0 + 1], b1[4 + k0 + 1]), 0.0f);

    // B (4x16): VGPR j holds row K = j + 2*half, column N = m; pad N>=2 with 0
    bool live = (m < 2);
    B0.x = live ? W2[(0 + k0) * 2 + m] : 0.0f;
    B0.y = live ? W2[(1 + k0) * 2 + m] : 0.0f;
    B1.x = live ? W2[(4 + k0) * 2 + m] : 0.0f;
    B1.y = live ? W2[(5 + k0) * 2 + m] : 0.0f;

    v8f c = {};
    c = __builtin_amdgcn_wmma_f32_16x16x4_f32(false, A0, false, B0, (short)0, c, false, false);
    c = __builtin_amdgcn_wmma_f32_16x16x4_f32(false, A1, false, B1, (short)0, c, false, false);

    if (m < 2) {                            // lanes 0,1,16,17 hold the useful columns
        #pragma unroll
        for (int v = 0; v < 8; ++v) {
            int node = base + 8 * half + v;
            if (node < N) {
                float val = dis[node] * c[v];   // p = dis * g
                p[2 * node + m]  = val;
                S2[2 * node + m] = val;         // self-loop init of accumulator
            }
        }
    }
}

// ---------------------------------------------------------------------------
// Pass 2: S2[dst,c] += p[src,c]  (2 atomics/edge, float2 gather)
// ---------------------------------------------------------------------------
__global__ void k_edge2(const int* __restrict__ src, const int* __restrict__ dst,
                        const float* __restrict__ p, float* __restrict__ S2, int E) {
    int t = blockIdx.x * blockDim.x + threadIdx.x;
    int e0 = t * 4;
    if (e0 + 3 < E) {
        int4 s = *(const int4*)(src + e0);
        int4 d = *(const int4*)(dst + e0);
        float2 p0 = *(const float2*)(p + 2 * s.x);
        float2 p1 = *(const float2*)(p + 2 * s.y);
        float2 p2 = *(const float2*)(p + 2 * s.z);
        float2 p3 = *(const float2*)(p + 2 * s.w);
        unsafeAtomicAdd(&S2[2 * d.x + 0], p0.x);
        unsafeAtomicAdd(&S2[2 * d.x + 1], p0.y);
        unsafeAtomicAdd(&S2[2 * d.y + 0], p1.x);
        unsafeAtomicAdd(&S2[2 * d.y + 1], p1.y);
        unsafeAtomicAdd(&S2[2 * d.z + 0], p2.x);
        unsafeAtomicAdd(&S2[2 * d.z + 1], p2.y);
        unsafeAtomicAdd(&S2[2 * d.w + 0], p3.x);
        unsafeAtomicAdd(&S2[2 * d.w + 1], p3.y);
    } else {
        for (int e = e0; e < E; ++e) {
            int s = src[e], d = dst[e];
            unsafeAtomicAdd(&S2[2 * d + 0], p[2 * s + 0]);
            unsafeAtomicAdd(&S2[2 * d + 1], p[2 * s + 1]);
        }
    }
}

// ---------------------------------------------------------------------------
// out[i,c] = dis[i]*S2[i,c] + b2[c]
// ---------------------------------------------------------------------------
__global__ void k_finalize(const float* __restrict__ dis, const float* __restrict__ S2,
                           const float* __restrict__ b2, float* __restrict__ out, int N) {
    int i = blockIdx.x * blockDim.x + threadIdx.x;
    if (i < N) {
        float  r = dis[i];
        float2 s = *(const float2*)(S2 + 2 * i);
        float2 o;
        o.x = fmaf(r, s.x, b2[0]);
        o.y = fmaf(r, s.y, b2[1]);
        *(float2*)(out + 2 * i) = o;
    }
}

extern "C" void kernel_launch(void* const* d_in, const int* in_sizes, int n_in,
                              void* d_out, int out_size, void* d_ws, size_t ws_size,
                              hipStream_t stream) {
    const float* x  = (const float*)d_in[0];
    const int*   ei = (const int*)d_in[1];   // [2, E] row-major; int per harness contract
    const float* W1 = (const float*)d_in[2]; // 8
    const float* b1 = (const float*)d_in[3]; // 8
    const float* W2 = (const float*)d_in[4]; // 8x2 row-major
    const float* b2 = (const float*)d_in[5]; // 2
    float*       out = (float*)d_out;

    const int N = in_sizes[0];
    const int E = in_sizes[1] / 2;
    const int* src = ei;
    const int* dst = ei + E;

    // Workspace layout (floats): deg/S1 | dis | q | p[2N] | S2[2N]  = 7N floats
    float* ws    = (float*)d_ws;
    float* degS1 = ws;
    float* dis   = ws + (size_t)N;
    float* q     = ws + 2 * (size_t)N;
    float* p     = ws + 3 * (size_t)N;
    float* S2    = ws + 5 * (size_t)N;

    const int BT = 256;                              // 8 wave32s per block
    int gN  = (N + BT - 1) / BT;
    int E4  = (E + 3) / 4;
    int gE  = (E4 + BT - 1) / BT;
    int gW  = (N + 127) / 128;                       // 128 nodes per block (16/wave)

    k_init_deg  <<<gN, BT, 0, stream>>>(degS1, N);
    k_deg_count <<<gE, BT, 0, stream>>>(dst, degS1, E);
    k_node_q    <<<gN, BT, 0, stream>>>(degS1, dis, q, x, N);
    k_edge1     <<<gE, BT, 0, stream>>>(src, dst, q, degS1, E);
    k_node_wmma <<<gW, BT, 0, stream>>>(dis, degS1, W1, b1, W2, p, S2, N);
    k_edge2     <<<gE, BT, 0, stream>>>(src, dst, p, S2, E);
    k_finalize  <<<gN, BT, 0, stream>>>(dis, S2, b2, out, N);
}